// SSMKernel_83588653515371
// MI455X (gfx1250) — compile-verified
//
#include <hip/hip_runtime.h>

// Problem dims (fixed by the reference)
#define BATCH   2
#define SEQ     2048
#define DM      1024
#define NS      16
#define NCHUNK  32
#define LC      (SEQ / NCHUNK)   // 64 timesteps per chunk
#define DPB     256              // d-channels per block
#define THREADS 256              // 8 wave32s

__device__ __forceinline__ float softplus_f(float v) {
    // jax.nn.softplus = log1p(exp(v)); overflow-safe
    return v > 20.0f ? v : log1pf(__expf(v));
}

// ---------------------------------------------------------------------------
// Chunk scan kernel. FINAL=false: pass 1 (emit P = prod(a), S = local state).
// FINAL=true : pass 3 (start from H0, emit y + D*x).
// Grid: BATCH * NCHUNK * (DM/DPB) = 256 blocks of 256 threads (2048 wave32s).
// ---------------------------------------------------------------------------
template <bool FINAL>
__global__ __launch_bounds__(THREADS)
void ssm_chunk_kernel(const float* __restrict__ x,
                      const float* __restrict__ Bm,
                      const float* __restrict__ Cm,
                      const float* __restrict__ delta,
                      const float* __restrict__ A_log,
                      const float* __restrict__ Dp,
                      const float* __restrict__ H0,
                      float* __restrict__ P,
                      float* __restrict__ S,
                      float* __restrict__ out)
{
    __shared__ __align__(16) float sB[LC * NS];   // 4 KB, contiguous in global
    __shared__ __align__(16) float sC[LC * NS];   // 4 KB (pass 3 only)

    const int tid  = threadIdx.x;
    const int blk  = blockIdx.x;
    const int dblk = blk & 3;                     // DM/DPB = 4
    const int c    = (blk >> 2) & (NCHUNK - 1);   // chunk id
    const int b    = blk >> 7;                    // batch id
    const int d    = dblk * DPB + tid;

    // ---- CDNA5 async-tensor path: stage B (and C for pass 3) into LDS -----
    {
        const float* gB = Bm + (size_t)(b * SEQ + c * LC) * NS;
        // low 32 bits of a generic LDS pointer == LDS byte offset (aperture)
        unsigned ldsB = (unsigned)(size_t)(&sB[0]) + (unsigned)tid * 16u;
        unsigned voff = (unsigned)tid * 16u;      // 256 threads * 16B = 4 KB
        asm volatile("global_load_async_to_lds_b128 %0, %1, %2 offset:0"
                     :: "v"(ldsB), "v"(voff),
                        "s"(reinterpret_cast<unsigned long long>(gB))
                     : "memory");
        if (FINAL) {
            const float* gC = Cm + (size_t)(b * SEQ + c * LC) * NS;
            unsigned ldsC = (unsigned)(size_t)(&sC[0]) + (unsigned)tid * 16u;
            asm volatile("global_load_async_to_lds_b128 %0, %1, %2 offset:0"
                         :: "v"(ldsC), "v"(voff),
                            "s"(reinterpret_cast<unsigned long long>(gC))
                         : "memory");
        }
        asm volatile("s_wait_asynccnt 0x0" ::: "memory");
    }
    __syncthreads();

    // ---- per-lane constants: A row = -exp(A_log[d, :]) --------------------
    float a_row[NS], h[NS], p[NS];
    {
        const float4* ar = reinterpret_cast<const float4*>(A_log + (size_t)d * NS);
        #pragma unroll
        for (int k = 0; k < 4; ++k) {
            float4 v = ar[k];
            a_row[4 * k + 0] = -__expf(v.x);
            a_row[4 * k + 1] = -__expf(v.y);
            a_row[4 * k + 2] = -__expf(v.z);
            a_row[4 * k + 3] = -__expf(v.w);
        }
    }
    #pragma unroll
    for (int n = 0; n < NS; ++n) { h[n] = 0.0f; p[n] = 1.0f; }
    if (FINAL) {
        #pragma unroll
        for (int n = 0; n < NS; ++n)
            h[n] = H0[((size_t)(b * NCHUNK + c) * NS + n) * DM + d];
    }
    const float dpar = FINAL ? Dp[d] : 0.0f;

    // ---- sequential scan over the chunk -----------------------------------
    const size_t xbase = ((size_t)b * SEQ + (size_t)c * LC) * DM + d;
    for (int t = 0; t < LC; ++t) {
        const size_t gi = xbase + (size_t)t * DM;
        const float xt = x[gi];
        const float dl = delta[gi];
        if (t + 8 < LC) {                          // global_prefetch_b8 stream
            __builtin_prefetch(&x[gi + 8 * DM], 0, 0);
            __builtin_prefetch(&delta[gi + 8 * DM], 0, 0);
        }
        const float dt = softplus_f(dl);
        float yt = 0.0f;
        #pragma unroll
        for (int n = 0; n < NS; ++n) {
            const float dA    = dt * a_row[n];
            const float abar  = __expf(fminf(dA, 80.0f));
            const float adA   = fabsf(dA);
            const float sgn   = (dA >= 0.0f) ? 1.0f : -1.0f;
            const float taylr = 1.0f + dA * (0.5f + dA * (1.0f / 6.0f));
            const float exact = (abar - 1.0f) / (dA + 1e-6f * sgn);
            const float scal  = (adA < 1e-4f) ? taylr : exact;
            const float bx    = dt * scal * sB[t * NS + n] * xt;
            h[n] = fmaf(abar, h[n], bx);
            if (FINAL) yt = fmaf(sC[t * NS + n], h[n], yt);
            else       p[n] *= abar;
        }
        if (FINAL) out[gi] = fmaf(dpar, xt, yt);
    }

    if (!FINAL) {
        #pragma unroll
        for (int n = 0; n < NS; ++n) {
            const size_t o = ((size_t)(b * NCHUNK + c) * NS + n) * DM + d;
            P[o] = p[n];
            S[o] = h[n];
        }
    }
}

// ---------------------------------------------------------------------------
// Pass 2: combine chunk summaries. One thread per (b, n, d); 32 serial steps.
// H0[c] = state entering chunk c:  h <- S[c] + P[c]*h  (elementwise, diagonal)
// All dims are powers of two -> pure mask/shift decode (no i32 div chains).
// ---------------------------------------------------------------------------
__global__ __launch_bounds__(256)
void ssm_combine_kernel(const float* __restrict__ P,
                        const float* __restrict__ S,
                        float* __restrict__ H0)
{
    const unsigned tid = blockIdx.x * blockDim.x + threadIdx.x; // [0, B*NS*DM)
    const unsigned d   = tid & (DM - 1);
    const unsigned n   = (tid >> 10) & (NS - 1);
    const unsigned b   = tid >> 14;
    float h = 0.0f;
    #pragma unroll 4
    for (int c = 0; c < NCHUNK; ++c) {
        const size_t o = ((size_t)((b * NCHUNK + c) * NS + n) << 10) + d;
        H0[o] = h;
        h = fmaf(P[o], h, S[o]);
    }
}

// ---------------------------------------------------------------------------
extern "C" void kernel_launch(void* const* d_in, const int* in_sizes, int n_in,
                              void* d_out, int out_size, void* d_ws, size_t ws_size,
                              hipStream_t stream)
{
    const float* x     = (const float*)d_in[0];
    const float* Bm    = (const float*)d_in[1];
    const float* Cm    = (const float*)d_in[2];
    const float* delta = (const float*)d_in[3];
    const float* A_log = (const float*)d_in[4];
    const float* Dp    = (const float*)d_in[5];
    float* out = (float*)d_out;

    // workspace: P | S | H0, each BATCH*NCHUNK*NS*DM floats (4 MB) -> 12 MB
    const size_t seg = (size_t)BATCH * NCHUNK * NS * DM;
    float* P  = (float*)d_ws;
    float* S  = P + seg;
    float* H0 = S + seg;

    const dim3 grid1(BATCH * NCHUNK * (DM / DPB));  // 256 blocks

    ssm_chunk_kernel<false><<<grid1, THREADS, 0, stream>>>(
        x, Bm, Cm, delta, A_log, Dp, nullptr, P, S, nullptr);

    ssm_combine_kernel<<<(BATCH * NS * DM) / 256, 256, 0, stream>>>(P, S, H0);

    ssm_chunk_kernel<true><<<grid1, THREADS, 0, stream>>>(
        x, Bm, Cm, delta, A_log, Dp, H0, P, S, out);
}